// SimpleProgressNet_16947940950581
// MI455X (gfx1250) — compile-verified
//
#include <hip/hip_runtime.h>
#include <hip/hip_bf16.h>

typedef __attribute__((ext_vector_type(2))) float v2f;
typedef __attribute__((ext_vector_type(8))) float v8f;

#define NEG_INF (-__builtin_inff())

// ---------------------------------------------------------------------------
// Kernel 1: SPP adaptive max pool, levels (4,3,2,1) over 96x96 planes.
// One 96-thread block per (frame, channel) plane. Thread t = q*24 + fc:
//   owns float4 column-group fc (cols 4fc..4fc+3), rows r = q + 4i, i=0..23.
// Row-band indices i/6 (level 4) and i/8 (level 3) are compile-time under
// full unroll, so accumulators live in registers. Column-block is constant
// per thread (fc/6, fc/8) since 4-aligned groups never straddle 24/32.
// This kernel reads 170 MB of frames -> the whole pipeline's HBM roofline.
// ---------------------------------------------------------------------------
__global__ __launch_bounds__(96) void spp_kernel(const float* __restrict__ frames,
                                                 float* __restrict__ pooled) {
  const int plane = blockIdx.x;           // n*3 + c
  const int n  = plane / 3;
  const int ch = plane % 3;
  const float* __restrict__ src = frames + (size_t)plane * 96 * 96;

  const int t  = threadIdx.x;             // 0..95
  const int q  = t / 24;                  // row phase 0..3
  const int fc = t % 24;                  // float4 column group

  float m4[4] = {NEG_INF, NEG_INF, NEG_INF, NEG_INF};
  float m3[3] = {NEG_INF, NEG_INF, NEG_INF};

#pragma unroll
  for (int i = 0; i < 24; ++i) {
    const int r = q + 4 * i;
    const float4 v = *(const float4*)(src + r * 96 + fc * 4);
    const float mv = fmaxf(fmaxf(v.x, v.y), fmaxf(v.z, v.w));
    m4[i / 6] = fmaxf(m4[i / 6], mv);     // row band r/24 == i/6 (compile-time)
    m3[i / 8] = fmaxf(m3[i / 8], mv);     // row band r/32 == i/8 (compile-time)
  }

  __shared__ float l4[4][96];
  __shared__ float l3[3][96];
  __shared__ float cell4[16];

#pragma unroll
  for (int b = 0; b < 4; ++b) l4[b][t] = m4[b];
#pragma unroll
  for (int b = 0; b < 3; ++b) l3[b][t] = m3[b];
  __syncthreads();

  float* __restrict__ out = pooled + (size_t)n * 90;

  if (t < 16) {                           // level 4: cell (bi,bj)
    const int bi = t / 4, bj = t % 4;     // cols 24bj..24bj+23 -> fc in [6bj,6bj+6)
    float m = NEG_INF;
    for (int qq = 0; qq < 4; ++qq)
      for (int f = 0; f < 6; ++f)
        m = fmaxf(m, l4[bi][qq * 24 + bj * 6 + f]);
    cell4[t] = m;
    out[ch * 16 + bi * 4 + bj] = m;
  } else if (t < 25) {                    // level 3
    const int u = t - 16;
    const int bi = u / 3, bj = u % 3;     // cols 32bj.. -> fc in [8bj,8bj+8)
    float m = NEG_INF;
    for (int qq = 0; qq < 4; ++qq)
      for (int f = 0; f < 8; ++f)
        m = fmaxf(m, l3[bi][qq * 24 + bj * 8 + f]);
    out[48 + ch * 9 + bi * 3 + bj] = m;
  }
  __syncthreads();

  if (t < 4) {                            // level 2 from level-4 cells
    const int bi = t / 2, bj = t % 2;
    const float m = fmaxf(
        fmaxf(cell4[(2 * bi) * 4 + 2 * bj], cell4[(2 * bi) * 4 + 2 * bj + 1]),
        fmaxf(cell4[(2 * bi + 1) * 4 + 2 * bj], cell4[(2 * bi + 1) * 4 + 2 * bj + 1]));
    out[75 + ch * 4 + bi * 2 + bj] = m;
  } else if (t == 4) {                    // level 1
    float m = NEG_INF;
    for (int k = 0; k < 16; ++k) m = fmaxf(m, cell4[k]);
    out[87 + ch] = m;
  }
}

// ---------------------------------------------------------------------------
// Kernel 2: forecasted_pooled = pooled(1536x90) @ W(90x90)^T + bias
// One wave per 16x16 output tile using V_WMMA_F32_16X16X4_F32.
// ISA layouts (7.12.2): A lane holds M=lane%16, K=2*half+v;
//                       B lane holds N=lane%16, K=2*half+v;
//                       C/D: row = v + 8*half, col = lane%16.
// Branchless: main K-loop (k<=84, kk<=87<90) loads unconditionally; single
// tail iteration uses clamped address + register select (no exec branches).
// B columns >= 90 only feed D columns that are never stored, so their
// addresses are merely clamped (no zeroing needed).
// ---------------------------------------------------------------------------
__global__ __launch_bounds__(32) void forecast_gemm_kernel(
    const float* __restrict__ A,      // 1536 x 90
    const float* __restrict__ W,      // 90 x 90   (B[k][n] = W[n][k])
    const float* __restrict__ bias,   // 90
    float* __restrict__ Y) {          // 1536 x 90
  const int tm = blockIdx.x;          // 0..95
  const int tn = blockIdx.y;          // 0..5
  const int lane = threadIdx.x;
  const int half = lane >> 4;
  const int lm = lane & 15;
  const int kb = 2 * half;            // 0 or 2

  const int arow  = tm * 16 + lm;     // < 1536 always
  const int bcol0 = tn * 16 + lm;     // may exceed 89
  const int bcol  = (bcol0 < 90) ? bcol0 : 89;  // clamped address only

  const float* __restrict__ Ar = A + (size_t)arow * 90;
  const float* __restrict__ Wr = W + (size_t)bcol * 90;

  v8f acc = {};
#pragma unroll 1
  for (int k = 0; k <= 84; k += 4) {  // kk = k + kb + v <= 87 < 90: no guards
    v2f a, b;
#pragma unroll
    for (int v = 0; v < 2; ++v) {
      a[v] = Ar[k + kb + v];
      b[v] = Wr[k + kb + v];
    }
    acc = __builtin_amdgcn_wmma_f32_16x16x4_f32(false, a, false, b,
                                                (short)0, acc, false, false);
  }
  {                                   // tail k = 88: kk in {88,89,90,91}
    v2f a, b;
#pragma unroll
    for (int v = 0; v < 2; ++v) {
      const int kk = 88 + kb + v;
      const int kc = (kk < 90) ? kk : 89;        // safe address
      const float av = Ar[kc];
      const float bv = Wr[kc];
      a[v] = (kk < 90) ? av : 0.0f;              // v_cndmask, no branch
      b[v] = (kk < 90) ? bv : 0.0f;
    }
    acc = __builtin_amdgcn_wmma_f32_16x16x4_f32(false, a, false, b,
                                                (short)0, acc, false, false);
  }
#pragma unroll
  for (int v = 0; v < 8; ++v) {
    const int r = tm * 16 + v + 8 * half;
    if (bcol0 < 90) Y[(size_t)r * 90 + bcol0] = acc[v] + bias[bcol0];
  }
}

// ---------------------------------------------------------------------------
// Kernel 3: fused embed: H = relu(relu(X @ W1^T + b1) @ W2pad^T + b2)
// X: N x 90, W1: 16 x 90, W2: 8 x 16. One wave per 16-row tile. The 16x16
// intermediate bounces through LDS to convert the WMMA C/D layout into the
// A layout for the second WMMA. Same branchless K-tail scheme as kernel 2.
// fc7 output rows >= 8 feed never-stored D columns -> address clamp only.
// ---------------------------------------------------------------------------
__global__ __launch_bounds__(32) void embed_kernel(
    const float* __restrict__ X,      // N x 90
    const float* __restrict__ W1,     // 16 x 90
    const float* __restrict__ b1,     // 16
    const float* __restrict__ W2,     // 8 x 16
    const float* __restrict__ b2,     // 8
    float* __restrict__ H) {          // N x 8
  __shared__ float e[16 * 16];
  const int tm = blockIdx.x;
  const int lane = threadIdx.x;
  const int half = lane >> 4;
  const int lm = lane & 15;
  const int kb = 2 * half;

  const float* __restrict__ Xr = X + (size_t)(tm * 16 + lm) * 90;
  const float* __restrict__ W1r = W1 + (size_t)lm * 90;   // N dim exactly 16

  v8f acc = {};
#pragma unroll 1
  for (int k = 0; k <= 84; k += 4) {  // no guards needed
    v2f a, b;
#pragma unroll
    for (int v = 0; v < 2; ++v) {
      a[v] = Xr[k + kb + v];
      b[v] = W1r[k + kb + v];
    }
    acc = __builtin_amdgcn_wmma_f32_16x16x4_f32(false, a, false, b,
                                                (short)0, acc, false, false);
  }
  {                                   // tail k = 88
    v2f a, b;
#pragma unroll
    for (int v = 0; v < 2; ++v) {
      const int kk = 88 + kb + v;
      const int kc = (kk < 90) ? kk : 89;
      const float av = Xr[kc];
      const float bv = W1r[kc];
      a[v] = (kk < 90) ? av : 0.0f;
      b[v] = (kk < 90) ? bv : 0.0f;
    }
    acc = __builtin_amdgcn_wmma_f32_16x16x4_f32(false, a, false, b,
                                                (short)0, acc, false, false);
  }
#pragma unroll
  for (int v = 0; v < 8; ++v) {
    const int r = v + 8 * half;
    e[r * 16 + lm] = fmaxf(acc[v] + b1[lm], 0.0f);  // relu
  }
  __syncthreads();

  const int w2row = (lm < 8) ? lm : 7;              // clamped address only
  const float* __restrict__ W2r = W2 + (size_t)w2row * 16;

  v8f acc2 = {};
#pragma unroll
  for (int k = 0; k < 16; k += 4) {   // K = 16 exact: fully unconditional
    v2f a, b;
#pragma unroll
    for (int v = 0; v < 2; ++v) {
      const int kk = k + kb + v;
      a[v] = e[lm * 16 + kk];
      b[v] = W2r[kk];
    }
    acc2 = __builtin_amdgcn_wmma_f32_16x16x4_f32(false, a, false, b,
                                                 (short)0, acc2, false, false);
  }
#pragma unroll
  for (int v = 0; v < 8; ++v) {
    const int r = tm * 16 + v + 8 * half;
    if (lm < 8) H[(size_t)r * 8 + lm] = fmaxf(acc2[v] + b2[lm], 0.0f);
  }
}

// ---------------------------------------------------------------------------
// Kernel 4: 192-step LSTM scan (B=8, HID=4) + sigmoid heads.
// tid = b*16 + u over the 4H=16 gate units; state h/c in LDS.
// Main cell updates state; forecast cell uses updated state, discarded.
// ---------------------------------------------------------------------------
__device__ __forceinline__ float sigm(float x) { return 1.0f / (1.0f + expf(-x)); }

__global__ __launch_bounds__(128) void lstm_kernel(
    const float* __restrict__ emb,    // 1536 x 8 (row = b*192 + s)
    const float* __restrict__ femb,   // 1536 x 8
    const float* __restrict__ Wih,    // 16 x 8
    const float* __restrict__ Whh,    // 16 x 4
    const float* __restrict__ bih,    // 16
    const float* __restrict__ bhh,    // 16
    const float* __restrict__ w8,     // 1 x 4
    const float* __restrict__ b8,     // 1
    float* __restrict__ prog,         // 8 x 192
    float* __restrict__ fprog) {      // 8 x 192
  __shared__ float h[8][4], c[8][4], hf[8][4], g[128];
  const int tid = threadIdx.x;
  const int b = tid >> 4;             // batch 0..7
  const int u = tid & 15;             // gate unit 0..15

  float wih[8], whh[4];
#pragma unroll
  for (int j = 0; j < 8; ++j) wih[j] = Wih[u * 8 + j];
#pragma unroll
  for (int j = 0; j < 4; ++j) whh[j] = Whh[u * 4 + j];
  const float bias = bih[u] + bhh[u];
  const float w8r0 = w8[0], w8r1 = w8[1], w8r2 = w8[2], w8r3 = w8[3];
  const float b8r = b8[0];

  if (tid < 32) { h[tid >> 2][tid & 3] = 0.0f; c[tid >> 2][tid & 3] = 0.0f; }
  __syncthreads();

  for (int s = 0; s < 192; ++s) {
    // ---- main cell ----
    {
      const float* x = emb + (b * 192 + s) * 8;
      float gg = bias;
#pragma unroll
      for (int j = 0; j < 8; ++j) gg += wih[j] * x[j];
#pragma unroll
      for (int j = 0; j < 4; ++j) gg += whh[j] * h[b][j];
      g[tid] = gg;
    }
    __syncthreads();
    if (tid < 32) {
      const int b2 = tid >> 2, u2 = tid & 3;
      const float i_ = sigm(g[b2 * 16 + u2]);
      const float f_ = sigm(g[b2 * 16 + 4 + u2]);
      const float g_ = tanhf(g[b2 * 16 + 8 + u2]);
      const float o_ = sigm(g[b2 * 16 + 12 + u2]);
      const float c2 = f_ * c[b2][u2] + i_ * g_;
      c[b2][u2] = c2;
      h[b2][u2] = o_ * tanhf(c2);
    }
    __syncthreads();
    if (tid < 8) {
      const float d = b8r + w8r0 * h[tid][0] + w8r1 * h[tid][1] +
                      w8r2 * h[tid][2] + w8r3 * h[tid][3];
      prog[tid * 192 + s] = sigm(d);
    }
    // ---- forecast cell (state not carried) ----
    {
      const float* xf = femb + (b * 192 + s) * 8;
      float gg = bias;
#pragma unroll
      for (int j = 0; j < 8; ++j) gg += wih[j] * xf[j];
#pragma unroll
      for (int j = 0; j < 4; ++j) gg += whh[j] * h[b][j];
      g[tid] = gg;
    }
    __syncthreads();
    if (tid < 32) {
      const int b2 = tid >> 2, u2 = tid & 3;
      const float i_ = sigm(g[b2 * 16 + u2]);
      const float f_ = sigm(g[b2 * 16 + 4 + u2]);
      const float g_ = tanhf(g[b2 * 16 + 8 + u2]);
      const float o_ = sigm(g[b2 * 16 + 12 + u2]);
      const float cf = f_ * c[b2][u2] + i_ * g_;
      hf[b2][u2] = o_ * tanhf(cf);
    }
    __syncthreads();
    if (tid < 8) {
      const float d = b8r + w8r0 * hf[tid][0] + w8r1 * hf[tid][1] +
                      w8r2 * hf[tid][2] + w8r3 * hf[tid][3];
      fprog[tid * 192 + s] = sigm(d);
    }
    __syncthreads();
  }
}

// ---------------------------------------------------------------------------
// Launcher
// ---------------------------------------------------------------------------
extern "C" void kernel_launch(void* const* d_in, const int* in_sizes, int n_in,
                              void* d_out, int out_size, void* d_ws, size_t ws_size,
                              hipStream_t stream) {
  (void)in_sizes; (void)n_in; (void)out_size; (void)ws_size;
  const float* frames        = (const float*)d_in[0];   // (8,192,3,96,96)
  const float* spp_fc_w      = (const float*)d_in[1];   // (16,90)
  const float* spp_fc_b      = (const float*)d_in[2];   // (16,)
  const float* forecast_fc_w = (const float*)d_in[3];   // (90,90)
  const float* forecast_fc_b = (const float*)d_in[4];   // (90,)
  const float* fc7_w         = (const float*)d_in[5];   // (8,16)
  const float* fc7_b         = (const float*)d_in[6];   // (8,)
  const float* lstm_wih      = (const float*)d_in[7];   // (16,8)
  const float* lstm_whh      = (const float*)d_in[8];   // (16,4)
  const float* lstm_bih      = (const float*)d_in[9];   // (16,)
  const float* lstm_bhh      = (const float*)d_in[10];  // (16,)
  const float* fc8_w         = (const float*)d_in[11];  // (1,4)
  const float* fc8_b         = (const float*)d_in[12];  // (1,)

  float* out = (float*)d_out;
  float* prog    = out;                        // 1536
  float* fprog   = out + 1536;                 // 1536
  float* pooled  = out + 3072;                 // 1536*90
  float* fpooled = out + 3072 + 1536 * 90;     // 1536*90

  float* emb  = (float*)d_ws;                  // 1536*8
  float* femb = emb + 1536 * 8;                // 1536*8

  // 1) SPP pooling (HBM-bound: reads 170 MB of frames)
  spp_kernel<<<dim3(4608), dim3(96), 0, stream>>>(frames, pooled);

  // 2) forecast GEMM (WMMA f32 16x16x4)
  forecast_gemm_kernel<<<dim3(96, 6), dim3(32), 0, stream>>>(
      pooled, forecast_fc_w, forecast_fc_b, fpooled);

  // 3) embed both streams (fused 2-stage WMMA MLP)
  embed_kernel<<<dim3(96), dim3(32), 0, stream>>>(
      pooled, spp_fc_w, spp_fc_b, fc7_w, fc7_b, emb);
  embed_kernel<<<dim3(96), dim3(32), 0, stream>>>(
      fpooled, spp_fc_w, spp_fc_b, fc7_w, fc7_b, femb);

  // 4) sequential LSTM scan + heads
  lstm_kernel<<<dim3(1), dim3(128), 0, stream>>>(
      emb, femb, lstm_wih, lstm_whh, lstm_bih, lstm_bhh,
      fc8_w, fc8_b, prog, fprog);
}